// Cell_57329223467782
// MI455X (gfx1250) — compile-verified
//
#include <hip/hip_runtime.h>
#include <math.h>
#include <stdint.h>

// ---------------------------------------------------------------------------
// SPDNet cell for MI455X (gfx1250).
// One workgroup (288 threads = 9 wave32 waves) per (batch, channel) pair.
// Whole chain resident in LDS (12 x 48x48 f32 matrices ~ 108 KB of the 320 KB
// WGP pool). Dense 48x48 matmuls via V_WMMA_F32_16X16X4_F32, one wave per
// 16x16 output tile. Eigendecompositions via parallel cyclic Jacobi with
// wave-local angle broadcast (__shfl, wave32) -> 2 barriers/round.
// Weight matrices staged asynchronously with the Tensor Data Mover
// (tensor_load_to_lds, 6-arg toolchain form), overlapped with the eigensolve,
// fenced by s_wait_tensorcnt.
// ---------------------------------------------------------------------------

typedef float v2f __attribute__((ext_vector_type(2)));
typedef float v8f __attribute__((ext_vector_type(8)));

#define D48        48
#define MAT_ELEMS  (48 * 48)          // 2304
#define NSWEEP     8
#define EPS_F      1e-5f
#define REIG_EPS_F 1e-4f

// LDS layout (element offsets into dynamic shared memory)
#define OFF_ST0 (0 * MAT_ELEMS)
#define OFF_ST1 (1 * MAT_ELEMS)
#define OFF_ST2 (2 * MAT_ELEMS)
#define OFF_ST3 (3 * MAT_ELEMS)
#define OFF_H1  (4 * MAT_ELEMS)
#define OFF_H2  (5 * MAT_ELEMS)
#define OFF_AW  (6 * MAT_ELEMS)
#define OFF_VV  (7 * MAT_ELEMS)
#define OFF_TM  (8 * MAT_ELEMS)
#define OFF_GS  (9 * MAT_ELEMS)
#define OFF_GI  (10 * MAT_ELEMS)
#define OFF_WL  (11 * MAT_ELEMS)      // staged weight matrix (TDM target)
#define OFF_FW  (12 * MAT_ELEMS)      // 48 eigenvalue f(w)
#define OFF_CS  (OFF_FW + D48)        // 24 rotation cosines
#define OFF_SN  (OFF_CS + 24)         // 24 rotation sines
#define SMEM_FLOATS (OFF_SN + 24)

#if defined(__has_builtin)
#if __has_builtin(__builtin_amdgcn_tensor_load_to_lds) && \
    __has_builtin(__builtin_amdgcn_s_wait_tensorcnt)
#define HAVE_TDM 1
#endif
#endif
#ifndef HAVE_TDM
#define HAVE_TDM 0
#endif

// ---- round-robin tournament pairing for parallel Jacobi (n=48) -------------
__device__ __forceinline__ void pair_pq(int r, int k, int& p, int& q) {
  const int m = 47;
  if (k == 0) { p = 47; q = r % m; }
  else        { p = (r + k) % m; q = (r + m - k) % m; }
}

// ---- 48x48 f32 matmul in LDS: smem[offR] = op(smem[offX]) * op(smem[offY]) -
// TA/TB are compile-time transpose flags -> straight-line DS addressing.
template <int TA, int TB>
__device__ __attribute__((noinline))
void mm48(int offR, int offX, int offY) {
  extern __shared__ float smem[];
  const int wave = threadIdx.x >> 5;    // 0..8 (uniform per wave)
  const int lane = threadIdx.x & 31;
  if (wave < 9) {
    const int ti = (wave / 3) * 16;
    const int tj = (wave % 3) * 16;
    const int m    = lane & 15;
    const int half = lane >> 4;         // K sub-pair selector (A/B lane split)
    const int row  = ti + m;
    const int col  = tj + m;
    const int aBase = TA ? (offX + (2 * half) * D48 + row)
                         : (offX + row * D48 + 2 * half);
    const int bBase = TB ? (offY + col * D48 + 2 * half)
                         : (offY + (2 * half) * D48 + col);
    v8f acc = {};
    #pragma unroll
    for (int k0 = 0; k0 < D48; k0 += 4) {
      v2f a, b;
      if (TA) { a.x = smem[aBase + k0 * D48]; a.y = smem[aBase + k0 * D48 + D48]; }
      else    { a.x = smem[aBase + k0];       a.y = smem[aBase + k0 + 1]; }
      if (TB) { b.x = smem[bBase + k0];       b.y = smem[bBase + k0 + 1]; }
      else    { b.x = smem[bBase + k0 * D48]; b.y = smem[bBase + k0 * D48 + D48]; }
      acc = __builtin_amdgcn_wmma_f32_16x16x4_f32(
          /*neg_a=*/false, a, /*neg_b=*/false, b,
          /*c_mod=*/(short)0, acc, /*reuse_a=*/false, /*reuse_b=*/false);
    }
    const int rBase = offR + (ti + 8 * half) * D48 + tj + m;
    #pragma unroll
    for (int r = 0; r < 8; ++r) smem[rBase + r * D48] = acc[r];
  }
  __syncthreads();
}

// ---- parallel cyclic Jacobi eigensolver (destroys smem[offA]) --------------
// Row phase: waves 0..7 own 3 rotation pairs each; lane 0 computes the angle,
// broadcast in-wave via __shfl (wave32), stashed to LDS for the column phase.
// Only 2 workgroup barriers per round.
__device__ __attribute__((noinline))
void jacobi48(int offA, int offV) {
  extern __shared__ float smem[];
  const int tid  = threadIdx.x;
  const int wave = tid >> 5;
  const int lane = tid & 31;
  for (int e = tid; e < MAT_ELEMS; e += blockDim.x)
    smem[offV + e] = ((e / D48) == (e % D48)) ? 1.0f : 0.0f;
  __syncthreads();

  for (int sw = 0; sw < NSWEEP; ++sw) {
    for (int r = 0; r < 47; ++r) {
      // ---- row phase: A <- J^T A (pairs touch disjoint rows) ----
      if (wave < 8) {
        #pragma unroll
        for (int t = 0; t < 3; ++t) {
          const int k = wave * 3 + t;
          int p, q; pair_pq(r, k, p, q);
          float c = 1.0f, s = 0.0f;
          if (lane == 0) {
            const float app = smem[offA + p * D48 + p];
            const float aqq = smem[offA + q * D48 + q];
            const float apq = smem[offA + p * D48 + q];
            if (fabsf(apq) > 1e-12f) {
              const float theta = (aqq - app) / (2.0f * apq);
              const float tt = copysignf(1.0f, theta) /
                               (fabsf(theta) + sqrtf(theta * theta + 1.0f));
              c = rsqrtf(tt * tt + 1.0f);
              s = tt * c;
            }
            smem[OFF_CS + k] = c;   // for the column phase (after barrier)
            smem[OFF_SN + k] = s;
          }
          c = __shfl(c, 0, 32);     // wave32 broadcast, no block sync
          s = __shfl(s, 0, 32);
          for (int j = lane; j < D48; j += 32) {
            const float ap = smem[offA + p * D48 + j];
            const float aq = smem[offA + q * D48 + j];
            smem[offA + p * D48 + j] = c * ap - s * aq;
            smem[offA + q * D48 + j] = s * ap + c * aq;
          }
        }
      }
      __syncthreads();
      // ---- col phase: A <- A J ; V <- V J (disjoint cols) ----
      for (int e = tid; e < 24 * D48; e += blockDim.x) {
        const int k = e / D48, i = e % D48;
        int p, q; pair_pq(r, k, p, q);
        const float c = smem[OFF_CS + k], s = smem[OFF_SN + k];
        const float ap = smem[offA + i * D48 + p];
        const float aq = smem[offA + i * D48 + q];
        smem[offA + i * D48 + p] = c * ap - s * aq;
        smem[offA + i * D48 + q] = s * ap + c * aq;
        const float vp = smem[offV + i * D48 + p];
        const float vq = smem[offV + i * D48 + q];
        smem[offV + i * D48 + p] = c * vp - s * vq;
        smem[offV + i * D48 + q] = s * vp + c * vq;
      }
      __syncthreads();
    }
  }
}

// ---- Tensor Data Mover: async stage 48x48 f32 weight -> OFF_WL -------------
#if HAVE_TDM
typedef unsigned int u32x4 __attribute__((ext_vector_type(4)));
typedef int i32x8 __attribute__((ext_vector_type(8)));
typedef int i32x4 __attribute__((ext_vector_type(4)));

__device__ __forceinline__ void tdm_issue_w(const float* __restrict__ Wg) {
  if (threadIdx.x < 32) {   // single wave issues the DMA (EXEC ignored by TDM)
    const unsigned long long ga = (unsigned long long)(uintptr_t)Wg;
    u32x4 g0;
    g0[0] = 1u;                                              // count=1 (user D#)
    g0[1] = (unsigned)(OFF_WL * sizeof(float));              // lds_addr (bytes)
    g0[2] = (unsigned)(ga & 0xFFFFFFFFull);                  // global_addr[31:0]
    g0[3] = (unsigned)((ga >> 32) & 0x01FFFFFFull) | (2u << 30); // [56:32]|type=2
    i32x8 g1;
    g1[0] = (int)(2u << 16);        // workgroup_mask=0, data_size=2 (4 bytes)
    g1[1] = (int)(48u << 16);       // tensor_dim0 = 48  (bits 79:48, low half)
    g1[2] = (int)(48u << 16);       // tensor_dim1 = 48  (bits 111:80, low half)
    g1[3] = (int)(48u << 16);       // tile_dim0   = 48  (bits 127:112)
    g1[4] = (int)48;                // tile_dim1   = 48  (bits 143:128)
    g1[5] = (int)48;                // tensor_dim0_stride = 48 (bits 207:160)
    g1[6] = 0;
    g1[7] = 0;
    const i32x4 z4 = {0, 0, 0, 0};          // groups 2/3 unused (2D tensor)
    const i32x8 z8 = {0, 0, 0, 0, 0, 0, 0, 0};
    __builtin_amdgcn_tensor_load_to_lds(g0, g1, z4, z4, z8, 0);
  }
}

__device__ __forceinline__ void tdm_wait_w() {
  if (threadIdx.x < 32) __builtin_amdgcn_s_wait_tensorcnt(0);
  __syncthreads();                  // publish WL to all waves
}
#endif

// ---- elementwise helpers (inlined; all end with a barrier) -----------------
__device__ __forceinline__ void g2lds(int dstOff, const float* __restrict__ g) {
  extern __shared__ float smem[];
  const float4* __restrict__ gv = (const float4*)g;
  float4* dv = (float4*)(smem + dstOff);
  for (int e = threadIdx.x; e < MAT_ELEMS / 4; e += blockDim.x) dv[e] = gv[e];
  __syncthreads();
}

__device__ __forceinline__ void lds2g(float* __restrict__ g, int srcOff) {
  extern __shared__ float smem[];
  const float4* sv = (const float4*)(smem + srcOff);
  float4* __restrict__ gv = (float4*)g;
  for (int e = threadIdx.x; e < MAT_ELEMS / 4; e += blockDim.x) gv[e] = sv[e];
  __syncthreads();
}

__device__ __forceinline__ void copy_lds(int dstOff, int srcOff) {
  extern __shared__ float smem[];
  const float4* sv = (const float4*)(smem + srcOff);
  float4* dv = (float4*)(smem + dstOff);
  for (int e = threadIdx.x; e < MAT_ELEMS / 4; e += blockDim.x) dv[e] = sv[e];
  __syncthreads();
}

__device__ __forceinline__ void avg_half(int dstOff, int aOff, int bOff) {
  extern __shared__ float smem[];
  const float4* av = (const float4*)(smem + aOff);
  const float4* bv = (const float4*)(smem + bOff);
  float4* dv = (float4*)(smem + dstOff);
  for (int e = threadIdx.x; e < MAT_ELEMS / 4; e += blockDim.x) {
    const float4 a = av[e], b = bv[e];
    float4 d;
    d.x = 0.5f * (a.x + b.x); d.y = 0.5f * (a.y + b.y);
    d.z = 0.5f * (a.z + b.z); d.w = 0.5f * (a.w + b.w);
    dv[e] = d;
  }
  __syncthreads();
}

// f(eigenvalues) from diag(smem[offA]): 0=ReEig clamp 1=sqrt 2=rsqrt 3=log 4=exp
__device__ __forceinline__ void make_fw(int offA, int mode) {
  extern __shared__ float smem[];
  const int t = threadIdx.x;
  if (t < D48) {
    const float w = smem[offA + t * D48 + t];
    float v;
    switch (mode) {
      case 0:  v = fmaxf(w, REIG_EPS_F);    break;
      case 1:  v = sqrtf(fmaxf(w, EPS_F));  break;
      case 2:  v = rsqrtf(fmaxf(w, EPS_F)); break;
      case 3:  v = logf(fmaxf(w, EPS_F));   break;
      default: v = expf(w);                 break;
    }
    smem[OFF_FW + t] = v;
  }
  __syncthreads();
}

// dst = V * diag(fw); a float4 never crosses a 48-col row boundary (4 | 48)
__device__ __forceinline__ void scale_cols(int dstOff, int vOff) {
  extern __shared__ float smem[];
  const float4* vv = (const float4*)(smem + vOff);
  float4* dv = (float4*)(smem + dstOff);
  for (int e = threadIdx.x; e < MAT_ELEMS / 4; e += blockDim.x) {
    const int cb = (4 * e) % D48;
    const float4 v = vv[e];
    const float4 f = *(const float4*)(smem + OFF_FW + cb);
    float4 d;
    d.x = v.x * f.x; d.y = v.y * f.y; d.z = v.z * f.z; d.w = v.w * f.w;
    dv[e] = d;
  }
  __syncthreads();
}

__device__ __forceinline__ void symmetrize(int offX) {
  extern __shared__ float smem[];
  for (int e = threadIdx.x; e < MAT_ELEMS; e += blockDim.x) {
    const int i = e / D48, j = e % D48;
    if (i < j) {
      const float t = 0.5f * (smem[offX + i * D48 + j] + smem[offX + j * D48 + i]);
      smem[offX + i * D48 + j] = t;
      smem[offX + j * D48 + i] = t;
    }
  }
  __syncthreads();
}

// ---- SPDNet primitive: out = W^T * ReEig(X) * W ----------------------------
// W is DMA'd into LDS by the TDM, overlapped with the whole eigensolve.
__device__ void spd_op(int offX, const float* __restrict__ Wg, int offOut) {
#if HAVE_TDM
  tdm_issue_w(Wg);                       // async W -> OFF_WL
#else
  g2lds(OFF_WL, Wg);
#endif
  copy_lds(OFF_AW, offX);
  jacobi48(OFF_AW, OFF_VV);
  make_fw(OFF_AW, 0);
  scale_cols(OFF_TM, OFF_VV);            // Tm = V * diag(max(w,eps))
  mm48<0, 1>(OFF_GI, OFF_TM, OFF_VV);    // Xr = Tm * V^T    (GI as temp)
#if HAVE_TDM
  tdm_wait_w();                          // s_wait_tensorcnt 0 + publish
#endif
  mm48<1, 0>(OFF_TM, OFF_WL, OFF_GI);    // Tm = W^T * Xr
  mm48<0, 0>(offOut, OFF_TM, OFF_WL);    // out = Tm * W
}

// ---- Karcher barycenter of (H1,H2); result left in OFF_AW ------------------
__device__ void bary2() {
  avg_half(OFF_AW, OFF_H1, OFF_H2);                 // G = (h1+h2)/2
  jacobi48(OFF_AW, OFF_VV);
  make_fw(OFF_AW, 1); scale_cols(OFF_TM, OFF_VV);
  mm48<0, 1>(OFF_GS, OFF_TM, OFF_VV);               // Gs = G^{1/2}
  make_fw(OFF_AW, 2); scale_cols(OFF_TM, OFF_VV);
  mm48<0, 1>(OFF_GI, OFF_TM, OFF_VV);               // Gi = G^{-1/2}
  // LA = logm(sym(Gi h1 Gi)) -> h1
  mm48<0, 0>(OFF_TM, OFF_GI, OFF_H1);
  mm48<0, 0>(OFF_AW, OFF_TM, OFF_GI);
  symmetrize(OFF_AW);
  jacobi48(OFF_AW, OFF_VV);
  make_fw(OFF_AW, 3); scale_cols(OFF_TM, OFF_VV);
  mm48<0, 1>(OFF_H1, OFF_TM, OFF_VV);
  // LB = logm(sym(Gi h2 Gi)) -> h2
  mm48<0, 0>(OFF_TM, OFF_GI, OFF_H2);
  mm48<0, 0>(OFF_AW, OFF_TM, OFF_GI);
  symmetrize(OFF_AW);
  jacobi48(OFF_AW, OFF_VV);
  make_fw(OFF_AW, 3); scale_cols(OFF_TM, OFF_VV);
  mm48<0, 1>(OFF_H2, OFF_TM, OFF_VV);
  // E = expm((LA+LB)/2) -> h1
  avg_half(OFF_AW, OFF_H1, OFF_H2);
  jacobi48(OFF_AW, OFF_VV);
  make_fw(OFF_AW, 4); scale_cols(OFF_TM, OFF_VV);
  mm48<0, 1>(OFF_H1, OFF_TM, OFF_VV);
  // result = sym(Gs E Gs) -> Aw
  mm48<0, 0>(OFF_TM, OFF_GS, OFF_H1);
  mm48<0, 0>(OFF_AW, OFF_TM, OFF_GS);
  symmetrize(OFF_AW);
}

// ---------------------------------------------------------------------------
__global__ __launch_bounds__(288)
void spdnet_cell_kernel(const float* __restrict__ s0, const float* __restrict__ s1,
                        const float* __restrict__ Wp0, const float* __restrict__ Wp1,
                        const float* __restrict__ Wops, float* __restrict__ out) {
  const int bc = blockIdx.x;            // 0..1023
  const int b  = bc >> 4;
  const int c  = bc & 15;

  // load input states into LDS
  g2lds(OFF_ST0, s0 + (size_t)bc * MAT_ELEMS);
  g2lds(OFF_ST1, s1 + (size_t)bc * MAT_ELEMS);

  // preprocess0 / preprocess1 (in place)
  spd_op(OFF_ST0, Wp0 + (size_t)c * MAT_ELEMS, OFF_ST0);
  spd_op(OFF_ST1, Wp1 + (size_t)c * MAT_ELEMS, OFF_ST1);

  const int stOff[4] = {OFF_ST0, OFF_ST1, OFF_ST2, OFF_ST3};
  const int idx1[4] = {0, 0, 1, 2};
  const int idx2[4] = {1, 1, 2, 3};

  for (int i = 0; i < 4; ++i) {
    const float* W1 = Wops + ((size_t)(2 * i)     * 16 + c) * MAT_ELEMS;
    const float* W2 = Wops + ((size_t)(2 * i + 1) * 16 + c) * MAT_ELEMS;
    if (threadIdx.x == 0) {             // gfx1250 global_prefetch_b8
      __builtin_prefetch(W2, 0, 3);
    }
    spd_op(stOff[idx1[i]], W1, OFF_H1);
    spd_op(stOff[idx2[i]], W2, OFF_H2);
    bary2();                            // result in OFF_AW

    // state (2+i) -> output channel-concat slot i
    lds2g(out + (((size_t)b * 64) + (size_t)i * 16 + c) * MAT_ELEMS, OFF_AW);
    if (i < 2) copy_lds(stOff[2 + i], OFF_AW);  // states 2,3 feed later steps
  }
}

// ---------------------------------------------------------------------------
extern "C" void kernel_launch(void* const* d_in, const int* in_sizes, int n_in,
                              void* d_out, int out_size, void* d_ws, size_t ws_size,
                              hipStream_t stream) {
  const float* s0   = (const float*)d_in[0];
  const float* s1   = (const float*)d_in[1];
  const float* Wp0  = (const float*)d_in[2];
  const float* Wp1  = (const float*)d_in[3];
  const float* Wops = (const float*)d_in[4];
  // d_in[5] = drop_prob (unused, == 0)
  float* out = (float*)d_out;

  const dim3 grid(64 * 16);   // one workgroup per (batch, channel) pair
  const dim3 block(288);      // 9 wave32 waves: one per 16x16 WMMA tile
  const size_t shmem = (size_t)SMEM_FLOATS * sizeof(float);

  hipLaunchKernelGGL(spdnet_cell_kernel, grid, block, shmem, stream,
                     s0, s1, Wp0, Wp1, Wops, out);
}